// QFCModel_21964462752052
// MI455X (gfx1250) — compile-verified
//
#include <hip/hip_runtime.h>
#include <hip/hip_bf16.h>

typedef float v2f __attribute__((ext_vector_type(2)));
typedef float v8f __attribute__((ext_vector_type(8)));
typedef int   v4i __attribute__((vector_size(4 * sizeof(int))));   // matches builtin param type

#define LDS_ROW 580                 // 576 floats + 4-float pad (bank-conflict-free A reads)
#define PROW    18                  // pooled row stride (conflict-free transpose)

#define AS1 __attribute__((address_space(1)))
#define AS3 __attribute__((address_space(3)))

#if __has_builtin(__builtin_amdgcn_global_load_async_to_lds_b128)
#define HAVE_ASYNC_LDS 1
#else
#pragma message("gfx1250: __builtin_amdgcn_global_load_async_to_lds_b128 NOT available; using float4 staging fallback")
#endif

__device__ __forceinline__ void wait_async_zero() {
#if __has_builtin(__builtin_amdgcn_s_wait_asynccnt)
    __builtin_amdgcn_s_wait_asynccnt(0);
#else
    asm volatile("s_wait_asynccnt 0x0" ::: "memory");
#endif
}

// ---------------- quantum-gate helpers (all in VGPRs, fully unrolled) ----------------

template<int MASK>
__device__ __forceinline__ void gate1(float* re, float* im,
    float u00r, float u00i, float u01r, float u01i,
    float u10r, float u10i, float u11r, float u11i)
{
#pragma unroll
    for (int s = 0; s < 16; ++s) {
        if (s & MASK) continue;
        const int s1 = s | MASK;
        float a0r = re[s],  a0i = im[s];
        float a1r = re[s1], a1i = im[s1];
        re[s]  = u00r*a0r - u00i*a0i + u01r*a1r - u01i*a1i;
        im[s]  = u00r*a0i + u00i*a0r + u01r*a1i + u01i*a1r;
        re[s1] = u10r*a0r - u10i*a0i + u11r*a1r - u11i*a1i;
        im[s1] = u10r*a0i + u10i*a0r + u11r*a1i + u11i*a1r;
    }
}

template<int MASK>
__device__ __forceinline__ void g_rx(float* re, float* im, float th) {
    float s, c; __sincosf(0.5f * th, &s, &c);
    gate1<MASK>(re, im, c, 0.f, 0.f, -s, 0.f, -s, c, 0.f);
}
template<int MASK>
__device__ __forceinline__ void g_ry(float* re, float* im, float th) {
    float s, c; __sincosf(0.5f * th, &s, &c);
    gate1<MASK>(re, im, c, 0.f, -s, 0.f, s, 0.f, c, 0.f);
}
template<int MASK>
__device__ __forceinline__ void g_rz(float* re, float* im, float th) {
    float s, c; __sincosf(0.5f * th, &s, &c);
    gate1<MASK>(re, im, c, -s, 0.f, 0.f, 0.f, 0.f, c, s);
}

// ---------------- kernel ----------------
// wire w lives at bit (3-w): wire0->8, wire1->4, wire2->2, wire3->1

__global__ __launch_bounds__(32)
void qfc_kernel(const float* __restrict__ x,
                const float* __restrict__ p_rx0,
                const float* __restrict__ p_ry0,
                const float* __restrict__ p_rz0,
                const float* __restrict__ p_crx0,
                float* __restrict__ out, int b)
{
    extern __shared__ float smem[];
    float* ldsX = smem;                       // 32 rows x 580 floats
    float* ldsP = smem + 32 * LDS_ROW;        // 32 rows x 18 floats (pooled transpose)

    const int lane  = threadIdx.x;            // 0..31, wave32
    const int mbase = blockIdx.x * 32;
    if (mbase + 32 > b) return;               // uniform guard (b % 32 == 0 in practice)

    // ---- stage 32x576 fp32 tile global -> LDS ----
    const float* gx = x + (size_t)mbase * 576;

    // keep the HBM stream ahead: touch the next block's tile (L2 prefetch)
    if (mbase + 64 <= b)
        __builtin_prefetch(gx + 32 * 576 + lane * 4, 0, 1);

#ifdef HAVE_ASYNC_LDS
    // fire-and-forget async DMA copies: 512 B per instruction, ASYNCcnt-tracked,
    // no VGPR staging, single wait before the GEMM.
    for (int j = 0; j < 144; ++j) {
        int g   = j * 128 + lane * 4;         // float offset within contiguous tile
        int row = g / 576;
        int col = g - row * 576;
        __builtin_amdgcn_global_load_async_to_lds_b128(
            (AS1 v4i*)(gx + g),
            (AS3 v4i*)(ldsX + row * LDS_ROW + col),
            0, 0);
    }
    wait_async_zero();
#else
#pragma unroll 4
    for (int j = 0; j < 144; ++j) {
        int g   = j * 128 + lane * 4;
        int row = g / 576;
        int col = g - row * 576;
        float4 v = *(const float4*)(gx + g);
        *(float4*)(ldsX + row * LDS_ROW + col) = v;
    }
#endif
    __syncthreads();

    // ---- pooling GEMM via V_WMMA_F32_16X16X4_F32 ----
    // A (16x4 f32): lanes 0-15 hold {K,K+1}, lanes 16-31 hold {K+2,K+3}  (ISA 7.12.2)
    // B (4x16 f32): C/D-mirrored striping: VGPR0 = {K (lanes<16), K+2 (lanes>=16)},
    //               VGPR1 = {K+1, K+3}, N = lane & 15
    const int nl = lane & 15;
    const int hi = (lane >> 4) & 1;
    const float* aptr0 = ldsX + nl * LDS_ROW + hi * 2;          // batch rows 0..15
    const float* aptr1 = aptr0 + 16 * LDS_ROW;                  // batch rows 16..31

    v8f acc0 = {0.f,0.f,0.f,0.f,0.f,0.f,0.f,0.f};
    v8f acc1 = {0.f,0.f,0.f,0.f,0.f,0.f,0.f,0.f};

    int kf = 0;                                // K offset in floats
    for (int rb = 0; rb < 4; ++rb) {           // pool-block row (by)
        const int nbase = rb * 4;
        for (int ri = 0; ri < 6; ++ri) {       // row within block
#pragma unroll
            for (int tc = 0; tc < 6; ++tc) {   // 4-column K chunk
                const int c0 = (4 * tc + 0) / 6;   // compile-time after unroll
                const int c1 = (4 * tc + 1) / 6;
                const int c2 = (4 * tc + 2) / 6;
                const int c3 = (4 * tc + 3) / 6;
                const int nA = nbase + (hi ? c2 : c0);
                const int nB = nbase + (hi ? c3 : c1);
                v2f bb;
                bb.x = (nl == nA) ? 1.0f : 0.0f;
                bb.y = (nl == nB) ? 1.0f : 0.0f;
                v2f a0 = *(const v2f*)(aptr0 + kf + tc * 4);
                v2f a1 = *(const v2f*)(aptr1 + kf + tc * 4);
                acc0 = __builtin_amdgcn_wmma_f32_16x16x4_f32(
                           false, a0, false, bb, (short)0, acc0, false, false);
                acc1 = __builtin_amdgcn_wmma_f32_16x16x4_f32(
                           false, a1, false, bb, (short)0, acc1, false, false);
            }
            kf += 24;
        }
    }

    // ---- transpose pooled sums through LDS: C layout -> per-lane angle rows ----
    __syncthreads();
#pragma unroll
    for (int i = 0; i < 8; ++i) {
        const int m = hi * 8 + i;                        // C/D: VGPR i -> rows i / i+8
        ldsP[m * PROW + nl]        = acc0[i];            // batch rows 0..15
        ldsP[(16 + m) * PROW + nl] = acc1[i];            // batch rows 16..31
    }
    __syncthreads();

    // ---- per-lane 4-qubit state simulation (1 batch element per lane) ----
    float ang[16];
#pragma unroll
    for (int q = 0; q < 16; ++q)
        ang[q] = ldsP[lane * PROW + q] * (1.0f / 36.0f); // mean of 6x6 block

    float re[16], im[16];
#pragma unroll
    for (int s = 0; s < 16; ++s) { re[s] = 0.f; im[s] = 0.f; }
    re[0] = 1.0f;

    // encoder
    g_rx<8>(re, im, ang[0]);  g_rx<4>(re, im, ang[1]);
    g_rx<2>(re, im, ang[2]);  g_rx<1>(re, im, ang[3]);
    g_ry<8>(re, im, ang[4]);  g_ry<4>(re, im, ang[5]);
    g_ry<2>(re, im, ang[6]);  g_ry<1>(re, im, ang[7]);
    g_rz<8>(re, im, ang[8]);  g_rz<4>(re, im, ang[9]);
    g_rz<2>(re, im, ang[10]); g_rz<1>(re, im, ang[11]);
    g_rx<8>(re, im, ang[12]); g_rx<4>(re, im, ang[13]);
    g_rx<2>(re, im, ang[14]); g_rx<1>(re, im, ang[15]);

    // trainable single-qubit gates (uniform scalars)
    g_rx<8>(re, im, p_rx0[0]);
    g_ry<4>(re, im, p_ry0[0]);
    g_rz<1>(re, im, p_rz0[0]);

    // CRX(theta) control wire0 (mask 8), target wire2 (mask 2)
    {
        float s, c; __sincosf(0.5f * p_crx0[0], &s, &c);
#pragma unroll
        for (int q = 0; q < 16; ++q) {
            if (!(q & 8) || (q & 2)) continue;
            const int q1 = q | 2;
            float a0r = re[q],  a0i = im[q];
            float a1r = re[q1], a1i = im[q1];
            re[q]  =  c * a0r + s * a1i;
            im[q]  =  c * a0i - s * a1r;
            re[q1] =  s * a0i + c * a1r;
            im[q1] = -s * a0r + c * a1i;
        }
    }

    // H on wire3 (mask 1)
    {
        const float r = 0.70710678118654752f;
        gate1<1>(re, im, r, 0.f, r, 0.f, r, 0.f, -r, 0.f);
    }
    // SX on wire2 (mask 2): 0.5*[[1+i,1-i],[1-i,1+i]]
    gate1<2>(re, im, 0.5f, 0.5f, 0.5f, -0.5f, 0.5f, -0.5f, 0.5f, 0.5f);

    // CNOT control wire3 (mask 1), target wire0 (mask 8)
#pragma unroll
    for (int q = 0; q < 16; ++q) {
        if (!(q & 1) || (q & 8)) continue;
        const int q1 = q | 8;
        float tr = re[q], ti = im[q];
        re[q] = re[q1]; im[q] = im[q1];
        re[q1] = tr;    im[q1] = ti;
    }

    // U12 on wires (1,2): when bit(wire1)=1 apply [[0,i],[-i,0]] on wire2
#pragma unroll
    for (int q = 0; q < 16; ++q) {
        if (!(q & 4) || (q & 2)) continue;
        const int q1 = q | 2;
        float xr = re[q],  xi = im[q];
        float yr = re[q1], yi = im[q1];
        re[q]  = -yi;  im[q]  =  yr;   //  i * amp(b=1)
        re[q1] =  xi;  im[q1] = -xr;   // -i * amp(b=0)
    }

    // ---- <Z_w> per wire, logits, log_softmax ----
    float z0 = 0.f, z1 = 0.f, z2 = 0.f, z3 = 0.f;
#pragma unroll
    for (int s = 0; s < 16; ++s) {
        float p = re[s]*re[s] + im[s]*im[s];
        z0 += (s & 8) ? -p : p;
        z1 += (s & 4) ? -p : p;
        z2 += (s & 2) ? -p : p;
        z3 += (s & 1) ? -p : p;
    }
    float l0 = z0 + z1;
    float l1 = z2 + z3;
    float mx  = fmaxf(l0, l1);
    float lse = mx + __logf(__expf(l0 - mx) + __expf(l1 - mx));

    float2 o = make_float2(l0 - lse, l1 - lse);
    *(float2*)(out + (size_t)(mbase + lane) * 2) = o;
}

// ---------------- host-side launch ----------------

extern "C" void kernel_launch(void* const* d_in, const int* in_sizes, int n_in,
                              void* d_out, int out_size, void* d_ws, size_t ws_size,
                              hipStream_t stream) {
    (void)n_in; (void)d_ws; (void)ws_size; (void)out_size;
    const float* x     = (const float*)d_in[0];
    const float* t_rx  = (const float*)d_in[1];
    const float* t_ry  = (const float*)d_in[2];
    const float* t_rz  = (const float*)d_in[3];
    const float* t_crx = (const float*)d_in[4];
    float* out = (float*)d_out;

    const int b = in_sizes[0] / 576;          // [b,1,24,24] fp32
    const int grid = (b + 31) / 32;           // 32 batch elements per wave32 block
    const size_t shmem = (size_t)(32 * LDS_ROW + 32 * PROW) * sizeof(float); // 76,544 B

    qfc_kernel<<<grid, 32, shmem, stream>>>(x, t_rx, t_ry, t_rz, t_crx, out, b);
}